// DirectionDPINN_D1_35304631173379
// MI455X (gfx1250) — compile-verified
//
#include <hip/hip_runtime.h>
#include <math.h>

#define EPSF 1e-8f

typedef __attribute__((ext_vector_type(2))) float v2f;
typedef __attribute__((ext_vector_type(8))) float v8f;

struct F3 { float x, y, z; };
__device__ __forceinline__ F3 operator+(F3 a, F3 b) { return {a.x + b.x, a.y + b.y, a.z + b.z}; }
__device__ __forceinline__ F3 operator*(float s, F3 a) { return {s * a.x, s * a.y, s * a.z}; }

// ---------------------------------------------------------------------------
// 256-element inclusive prefix-scan combine done with V_WMMA_F32_16X16X4_F32.
// Partials laid out as X[m][k] = p[16*m + k].
//   Part 1:  Y = X * U   (U[j][n] = j<=n)  -> row-wise inclusive scan
//            (4 chained K=4 WMMAs); lane n==15 spills row totals to LDS.
//   Part 2:  acc += L * W (L[m][j] = j<m, W[j][n] = rowTot[j])
//            -> adds exclusive row offsets; result = full 256 inclusive scan.
// VGPR layouts per CDNA5 ISA 7.12.2:
//   A 16x4 : lanes 0-15 -> j=0(v0),1(v1); lanes 16-31 -> j=2(v0),3(v1); M = lane&15
//   B 4x16 : lanes 0-15 -> rows 0(v0),1(v1); lanes 16-31 -> rows 2(v0),3(v1); N = lane&15
//   C/D    : VGPR r -> M = r + 8*(lane>=16); N = lane&15
// Executed by one full wave (EXEC all ones; wave-uniform guard outside).
// ---------------------------------------------------------------------------
__device__ __forceinline__ v8f scan256_part1(const float* __restrict__ src,
                                             float* __restrict__ rowT, int lane) {
  const int n = lane & 15;
  const int half = lane >> 4;
  v8f acc = {};
#pragma unroll
  for (int kk = 0; kk < 4; ++kk) {
    const int j0 = kk * 4 + 2 * half;   // global K index held in VGPR0 (VGPR1 = j0+1)
    v2f a, bb;
    a.x = src[n * 16 + j0];
    a.y = src[n * 16 + j0 + 1];
    bb.x = (j0 <= n) ? 1.0f : 0.0f;
    bb.y = (j0 + 1 <= n) ? 1.0f : 0.0f;
    acc = __builtin_amdgcn_wmma_f32_16x16x4_f32(false, a, false, bb,
                                                (short)0, acc, false, false);
  }
  if (n == 15) {
#pragma unroll
    for (int r = 0; r < 8; ++r) rowT[r + 8 * half] = acc[r];
  }
  return acc;
}

__device__ __forceinline__ void scan256_part2(v8f acc, const float* __restrict__ rowT,
                                              float* __restrict__ dst, int lane) {
  const int n = lane & 15;
  const int half = lane >> 4;
#pragma unroll
  for (int kk = 0; kk < 4; ++kk) {
    const int j0 = kk * 4 + 2 * half;
    v2f a, bb;
    a.x = (j0 < n) ? 1.0f : 0.0f;      // strict lower-triangular: A[m=n][j0]
    a.y = (j0 + 1 < n) ? 1.0f : 0.0f;
    bb.x = rowT[j0];                   // broadcast row totals across N
    bb.y = rowT[j0 + 1];
    acc = __builtin_amdgcn_wmma_f32_16x16x4_f32(false, a, false, bb,
                                                (short)0, acc, false, false);
  }
#pragma unroll
  for (int r = 0; r < 8; ++r) dst[(r + 8 * half) * 16 + n] = acc[r];
}

// ---------------------------------------------------------------------------
// 6D -> rotation matrix -> quaternion (literal port of the reference math).
// ---------------------------------------------------------------------------
__device__ __forceinline__ void quat_row(const float* __restrict__ s6, float* __restrict__ q) {
  const float r1x = s6[0], r1y = s6[1], r1z = s6[2];
  const float r2x = s6[3], r2y = s6[4], r2z = s6[5];
  const float n1 = sqrtf(r1x * r1x + r1y * r1y + r1z * r1z) + EPSF;
  const float b1x = r1x / n1, b1y = r1y / n1, b1z = r1z / n1;
  const float dot = b1x * r2x + b1y * r2y + b1z * r2z;
  const float ox = r2x - dot * b1x, oy = r2y - dot * b1y, oz = r2z - dot * b1z;
  const float n2 = sqrtf(ox * ox + oy * oy + oz * oz) + EPSF;
  const float b2x = ox / n2, b2y = oy / n2, b2z = oz / n2;
  const float b3x = b1y * b2z - b1z * b2y;
  const float b3y = b1z * b2x - b1x * b2z;
  const float b3z = b1x * b2y - b1y * b2x;
  // R columns are b1,b2,b3
  const float r00 = b1x, r01 = b2x, r02 = b3x;
  const float r10 = b1y, r11 = b2y, r12 = b3y;
  const float r20 = b1z, r21 = b2z, r22 = b3z;
  const float trace = r00 + r11 + r22;
  const float s1 = sqrtf(fmaxf(trace + 1.0f, EPSF)) * 2.0f;
  const float s2 = sqrtf(fmaxf(1.0f + r00 - r11 - r22, EPSF)) * 2.0f;
  const float s3 = sqrtf(fmaxf(1.0f + r11 - r00 - r22, EPSF)) * 2.0f;
  const float s4 = sqrtf(fmaxf(1.0f + r22 - r00 - r11, EPSF)) * 2.0f;
  const float q1[4] = {0.25f * s1, (r21 - r12) / s1, (r02 - r20) / s1, (r10 - r01) / s1};
  const float q2[4] = {(r21 - r12) / s2, 0.25f * s2, (r01 + r10) / s2, (r02 + r20) / s2};
  const float q3[4] = {(r02 - r20) / s3, (r01 + r10) / s3, 0.25f * s3, (r12 + r21) / s3};
  const float q4[4] = {(r10 - r01) / s4, (r02 + r20) / s4, (r12 + r21) / s4, 0.25f * s4};
  const bool m1 = trace > EPSF;
  const bool m2 = (r00 > r11) && (r00 > r22);
  const bool m3 = r11 > r22;
  float qq[4];
#pragma unroll
  for (int i = 0; i < 4; ++i)
    qq[i] = m1 ? q1[i] : (m2 ? q2[i] : (m3 ? q3[i] : q4[i]));
  const float nq = sqrtf(qq[0] * qq[0] + qq[1] * qq[1] + qq[2] * qq[2] + qq[3] * qq[3]) + EPSF;
  q[0] = qq[0] / nq; q[1] = qq[1] / nq; q[2] = qq[2] / nq; q[3] = qq[3] / nq;
}

// ---------------------------------------------------------------------------
// One 256-thread block per batch row b. Each thread owns CHUNK=16 RK4 steps.
// Phase 1: sequential chunk partials (Pv, Dt, h). Phase 2: WMMA 256-scans
// (v then z) by waves 0..2. Phase 3: re-stream chunk, emit v,z,quat rows.
// ---------------------------------------------------------------------------
__global__ __launch_bounds__(256) void dpinn_fused_kernel(
    const float* __restrict__ sixd, const float* __restrict__ accel,
    const float* __restrict__ v0g, const float* __restrict__ z0g,
    const float* __restrict__ tg, float* __restrict__ out, int N) {
  const int b = blockIdx.x;
  const int tid = (int)threadIdx.x;
  const int lane = tid & 31;
  const int wave = tid >> 5;
  const int steps = N - 1;
  const int k0 = tid * 16;
  const int ns = min(16, max(0, steps - k0));

  const float* tB = tg + (size_t)b * N;
  const float* aB = accel + (size_t)b * N * 3;
  const float* sB = sixd + (size_t)b * N * 6;
  float* oB = out + (size_t)b * N * 10;

  __shared__ float P[3][256];    // partials (reused for v then z round)
  __shared__ float Sc[3][256];   // inclusive-scan results (reused)
  __shared__ float rowT[3][16];  // per-component row-total scratch

  if (ns > 0) __builtin_prefetch(sB + (size_t)(k0 + 1) * 6, 0, 0);  // global_prefetch_b8

  // ---- Phase 1: chunk partials -------------------------------------------
  F3 S = {0, 0, 0}, h = {0, 0, 0};
  float Dt = 0.0f;
  {
    float tp = 0.0f;
    F3 ap = {0, 0, 0};
    if (ns > 0) {
      tp = tB[k0];
      ap = F3{aB[3 * k0], aB[3 * k0 + 1], aB[3 * k0 + 2]};
    }
#pragma unroll
    for (int k = 0; k < 16; ++k) {
      if (k < ns) {
        const int idx = k0 + k + 1;
        const float tn = tB[idx];
        const F3 an = {aB[3 * idx], aB[3 * idx + 1], aB[3 * idx + 2]};
        const float dt = tn - tp;
        const F3 amid = 0.5f * (ap + an);
        const F3 c = (dt * (1.0f / 6.0f)) * (ap + 4.0f * amid + an);        // dv
        const F3 g = ((dt * dt) * (1.0f / 6.0f)) * (ap + 2.0f * amid);      // dz - dt*v
        h = h + (dt * S + g);
        Dt += dt;
        S = S + c;
        tp = tn;
        ap = an;
      }
    }
  }

  P[0][tid] = S.x; P[1][tid] = S.y; P[2][tid] = S.z;
  __syncthreads();

  // ---- Phase 2a: v-scan (waves 0..2, one component each, WMMA) -----------
  v8f acc{};
  if (wave < 3) acc = scan256_part1(P[wave], rowT[wave], lane);
  __syncthreads();
  if (wave < 3) scan256_part2(acc, rowT[wave], Sc[wave], lane);
  __syncthreads();

  const F3 vb = {v0g[3 * b + 0] + (tid ? Sc[0][tid - 1] : 0.0f),
                 v0g[3 * b + 1] + (tid ? Sc[1][tid - 1] : 0.0f),
                 v0g[3 * b + 2] + (tid ? Sc[2][tid - 1] : 0.0f)};

  // ---- Phase 2b: z-scan --------------------------------------------------
  const F3 Pz = Dt * vb + h;
  __syncthreads();
  P[0][tid] = Pz.x; P[1][tid] = Pz.y; P[2][tid] = Pz.z;
  __syncthreads();
  if (wave < 3) acc = scan256_part1(P[wave], rowT[wave], lane);
  __syncthreads();
  if (wave < 3) scan256_part2(acc, rowT[wave], Sc[wave], lane);
  __syncthreads();

  const F3 zb = {z0g[3 * b + 0] + (tid ? Sc[0][tid - 1] : 0.0f),
                 z0g[3 * b + 1] + (tid ? Sc[1][tid - 1] : 0.0f),
                 z0g[3 * b + 2] + (tid ? Sc[2][tid - 1] : 0.0f)};

  // ---- Phase 3: re-stream chunk, emit fused 10-float rows ----------------
  {
    F3 v = vb, z = zb;
    float tp = 0.0f;
    F3 ap = {0, 0, 0};
    if (ns > 0) {
      tp = tB[k0];
      ap = F3{aB[3 * k0], aB[3 * k0 + 1], aB[3 * k0 + 2]};
    }
    for (int k = 0; k < ns; ++k) {
      const int idx = k0 + k + 1;
      const float tn = tB[idx];
      const F3 an = {aB[3 * idx], aB[3 * idx + 1], aB[3 * idx + 2]};
      const float dt = tn - tp;
      const F3 amid = 0.5f * (ap + an);
      const F3 zn = z + (dt * v + ((dt * dt) * (1.0f / 6.0f)) * (ap + 2.0f * amid));
      const F3 vn = v + (dt * (1.0f / 6.0f)) * (ap + 4.0f * amid + an);
      float q[4];
      quat_row(sB + (size_t)idx * 6, q);
      float2* o = (float2*)(oB + (size_t)idx * 10);  // 40B row stride -> 8B aligned
      o[0] = make_float2(vn.x, vn.y);
      o[1] = make_float2(vn.z, zn.x);
      o[2] = make_float2(zn.y, zn.z);
      o[3] = make_float2(q[0], q[1]);
      o[4] = make_float2(q[2], q[3]);
      v = vn;
      z = zn;
      tp = tn;
      ap = an;
    }
  }

  if (tid == 0) {  // row 0: initial state + quat(row 0)
    float q[4];
    quat_row(sB, q);
    float2* o = (float2*)oB;
    o[0] = make_float2(v0g[3 * b + 0], v0g[3 * b + 1]);
    o[1] = make_float2(v0g[3 * b + 2], z0g[3 * b + 0]);
    o[2] = make_float2(z0g[3 * b + 1], z0g[3 * b + 2]);
    o[3] = make_float2(q[0], q[1]);
    o[4] = make_float2(q[2], q[3]);
  }
}

extern "C" void kernel_launch(void* const* d_in, const int* in_sizes, int n_in,
                              void* d_out, int out_size, void* d_ws, size_t ws_size,
                              hipStream_t stream) {
  (void)n_in; (void)d_ws; (void)ws_size; (void)out_size;
  const float* sixd  = (const float*)d_in[0];
  const float* accel = (const float*)d_in[1];
  const float* v0    = (const float*)d_in[2];
  const float* z0    = (const float*)d_in[3];
  const float* t     = (const float*)d_in[4];
  float* out = (float*)d_out;

  const int B = in_sizes[2] / 3;        // v0 is (B,3)
  const int N = in_sizes[4] / B;        // t is (B,N,1)

  dim3 grid(B), block(256);
  hipLaunchKernelGGL(dpinn_fused_kernel, grid, block, 0, stream,
                     sixd, accel, v0, z0, t, out, N);
}